// ClusterSeedClsOffsetShift_32555852103941
// MI455X (gfx1250) — compile-verified
//
#include <hip/hip_runtime.h>
#include <stdint.h>

// ---------------- problem constants ----------------
#define HWN       (1u << 21)        // 1024*2048 pixels
#define IMG_W     2048
#define IMG_H     1024
#define NT        256               // threads per block (8 waves, wave32)
#define GRAIN     8                 // pixels per thread in strided kernels
#define TILE      4096              // pixels per block in the TDM-staged distance kernel
#define MIN_PIXEL       160
#define MIN_INST_PIXEL  160
#define THRESHOLD       0.5f
#define MIN_SEED_THRESH 0.5f
#define INST_RATIO      0.5f
#define DIST_LN2        0.69314718055994530942f   // dist=exp(-q)>0.5  <=>  q<ln2
#define MAX_ITERS 64

// ---------------- workspace scalar block ----------------
struct Scal {
  unsigned long long argmax_key;   // (score_bits<<32) | (0xFFFFFFFF - idx)
  int sum_uncl;
  int psum;
  int inter;
  int done;
  int count;
  int valid_count;
  int pad0;
  float c0, c1, s0, s1;
  int counts[68];                  // label histogram, labels in [0,65]
};

// ---------------- CDNA5 TDM: 1-D tile load, global -> LDS ----------------
typedef unsigned int v4u __attribute__((ext_vector_type(4)));
typedef int          v8i __attribute__((ext_vector_type(8)));
typedef int          v4i __attribute__((ext_vector_type(4)));
typedef float        v2f __attribute__((ext_vector_type(2)));
typedef float        v8f __attribute__((ext_vector_type(8)));

__device__ __forceinline__ void tdm_load_f32_1d(const float* gsrc,
                                                unsigned lds_byte_addr,
                                                int nelem) {
  unsigned long long ga = (unsigned long long)(uintptr_t)gsrc;
  v4u g0; v8i g1; v4i gz4; v8i gz8;
  gz4[0] = 0; gz4[1] = 0; gz4[2] = 0; gz4[3] = 0;
  gz8[0] = 0; gz8[1] = 0; gz8[2] = 0; gz8[3] = 0;
  gz8[4] = 0; gz8[5] = 0; gz8[6] = 0; gz8[7] = 0;
  // D# group 0: count=1 (valid user descriptor), lds_addr, global_addr[56:0], type=2
  g0[0] = 1u;
  g0[1] = lds_byte_addr;
  g0[2] = (unsigned)(ga & 0xFFFFFFFFull);
  g0[3] = (unsigned)((ga >> 32) & 0x1FFFFFFull) | (2u << 30);
  // D# group 1: data_size=2 (4B), tensor_dim0=nelem, tensor_dim1=1,
  //             tile_dim0=nelem, tile_dim1=1, tensor_dim0_stride=nelem
  g1[0] = (int)(2u << 16);
  g1[1] = (int)(((unsigned)nelem & 0xFFFFu) << 16);                 // tensor_dim0 lo16
  g1[2] = (int)((((unsigned)nelem >> 16) & 0xFFFFu) | (1u << 16));  // tensor_dim0 hi16 | tensor_dim1 lo16
  g1[3] = (int)(((unsigned)nelem & 0xFFFFu) << 16);                 // tensor_dim1 hi16(=0) | tile_dim0
  g1[4] = 1;                                                        // tile_dim1=1, tile_dim2=0
  g1[5] = nelem;                                                    // tensor_dim0_stride lo32
  g1[6] = 0;
  g1[7] = 0;
  __builtin_amdgcn_tensor_load_to_lds(g0, g1, gz4, gz4, gz8, 0);
}

// ---------------- CDNA5 WMMA wave32 sum: A(partials) x B(ones) ----------------
// A 16x4 f32: VGPR0 lane m = A[m][0], lane 16+m = A[m][2]; VGPR1 -> K=1,3 (zeros).
// D row m (any column) = partial(lane m) + partial(lane m+16); summing the 8 D
// VGPRs leaves the lane0-15 half-sum / lane16-31 half-sum; shfl_xor(16) finishes.
__device__ __forceinline__ float wave_sum_wmma(float x) {
  v2f a; a[0] = x;    a[1] = 0.0f;
  v2f b; b[0] = 1.0f; b[1] = 1.0f;   // all-ones B regardless of layout
  v8f c = {};
  v8f d = __builtin_amdgcn_wmma_f32_16x16x4_f32(false, a, false, b,
                                                (short)0, c, false, false);
  float acc = d[0] + d[1] + d[2] + d[3] + d[4] + d[5] + d[6] + d[7];
  acc += __shfl_xor(acc, 16, 32);
  return acc;
}

// ---------------- kernels ----------------

__global__ void k_zero(Scal* sc) {
  if (threadIdx.x == 0) {
    sc->argmax_key = 0ull; sc->sum_uncl = 0; sc->psum = 0; sc->inter = 0;
    sc->done = 0; sc->count = 1; sc->valid_count = 0;
  }
  if (threadIdx.x < 68) sc->counts[threadIdx.x] = 0;
}

// Pass 1: offsets, grid-sample shift, seed; init masks/labels; count valid.
__global__ __launch_bounds__(NT) void k_init(const float* __restrict__ pred,
                                             float* __restrict__ emb0,
                                             float* __restrict__ emb1,
                                             float* __restrict__ seedv,
                                             unsigned char* __restrict__ uncl,
                                             int* __restrict__ labels,
                                             Scal* sc) {
  __shared__ int red[NT];
  int j = blockIdx.x * NT + threadIdx.x;
  int y = j >> 11;           // /2048
  int x = j & 2047;
  float o0 = tanhf(pred[j]);
  float o1 = tanhf(pred[HWN + j]);
  float se0 = o0 + (float)x * (2.0f / 2047.0f);
  float se1 = o1 + (float)y * (1.0f / 1023.0f);
  float gx = 2.0f * (se0 * (1024.0f / 2047.0f) - 0.5f);
  float gy = 2.0f * (se1 * (1024.0f / 1023.0f) - 0.5f);
  // bilinear sample of tanh(pred[0:2]) at (gx,gy), zeros padding, align_corners=False
  float fx = (gx + 1.0f) * 0.5f * (float)IMG_W - 0.5f;
  float fy = (gy + 1.0f) * 0.5f * (float)IMG_H - 0.5f;
  float x0f = floorf(fx), y0f = floorf(fy);
  float wx = fx - x0f, wy = fy - y0f;
  int x0 = (int)x0f, y0 = (int)y0f;
  float a0 = 0.0f, a1 = 0.0f;
  #pragma unroll
  for (int dy = 0; dy < 2; ++dy) {
    #pragma unroll
    for (int dx = 0; dx < 2; ++dx) {
      int yi = y0 + dy, xi = x0 + dx;
      float w = (dy ? wy : 1.0f - wy) * (dx ? wx : 1.0f - wx);
      if (yi >= 0 && yi < IMG_H && xi >= 0 && xi < IMG_W) {
        int jj = yi * IMG_W + xi;
        a0 += w * tanhf(pred[jj]);
        a1 += w * tanhf(pred[HWN + jj]);
      }
    }
  }
  emb0[j] = se0 + a0;
  emb1[j] = se1 + a1;
  // seed = softmax(pred[5:7])[1] = sigmoid(p6 - p5)
  float sd = 1.0f / (1.0f + expf(pred[5u * HWN + j] - pred[6u * HWN + j]));
  seedv[j] = sd;
  int v = sd > MIN_SEED_THRESH;
  uncl[j] = (unsigned char)v;
  labels[j] = 0;
  red[threadIdx.x] = v; __syncthreads();
  for (int s = NT / 2; s > 0; s >>= 1) {
    if (threadIdx.x < s) red[threadIdx.x] += red[threadIdx.x + s];
    __syncthreads();
  }
  if (threadIdx.x == 0) atomicAdd(&sc->valid_count, red[0]);
}

__global__ void k_start(Scal* sc) {
  sc->done = (sc->valid_count > MIN_PIXEL) ? 0 : 1;
  sc->count = 1;
  sc->argmax_key = 0ull;
  sc->sum_uncl = 0;
  sc->psum = 0;
  sc->inter = 0;
}

// Fused apply(t-1) + reduce(t): apply previous proposal to labels/unclustered,
// then sum(unclustered) and argmax(seed*unclustered) with first-index tiebreak.
__global__ __launch_bounds__(NT) void k_ar(const float* __restrict__ seedv,
                                           unsigned char* __restrict__ uncl,
                                           const unsigned char* __restrict__ prop,
                                           int* __restrict__ labels,
                                           Scal* sc, int first) {
  __shared__ unsigned long long rk[NT];
  __shared__ int rs[NT];
  if (sc->done) return;                      // uniform: labels frozen once done
  int psum = sc->psum;
  float ratio = (float)sc->inter / (float)(psum > 1 ? psum : 1);
  int assign = (!first) && (psum > MIN_INST_PIXEL) && (ratio > INST_RATIO);
  int doApply = !first;
  int cnt = sc->count;
  int tid = threadIdx.x;
  int base = blockIdx.x * (NT * GRAIN);
  unsigned long long key = 0ull; int su = 0;
  #pragma unroll
  for (int k = 0; k < GRAIN; ++k) {
    int j = base + k * NT + tid;
    int u = uncl[j];
    if (doApply && prop[j]) {
      if (assign) labels[j] = cnt;
      if (u) { u = 0; uncl[j] = 0; }         // unclustered_s & ~proposal
    }
    unsigned fb = __float_as_uint(seedv[j]); // seed>0 => bits monotone
    unsigned long long kk =
        u ? (((unsigned long long)fb << 32) | (0xFFFFFFFFu - (unsigned)j)) : 0ull;
    su += u;
    key = (kk > key) ? kk : key;
  }
  rk[tid] = key; rs[tid] = su; __syncthreads();
  for (int s = NT / 2; s > 0; s >>= 1) {
    if (tid < s) {
      if (rk[tid + s] > rk[tid]) rk[tid] = rk[tid + s];
      rs[tid] += rs[tid + s];
    }
    __syncthreads();
  }
  if (tid == 0) {
    atomicAdd(&sc->sum_uncl, rs[0]);
    atomicMax(&sc->argmax_key, rk[0]);
  }
}

// Per-iteration scalar step: count++ for t-1, finalize done, fetch center/sigma,
// clear seed pixel, reset all accumulators for this iteration.
__global__ void k_mid(const float* __restrict__ pred,
                      const float* __restrict__ emb0,
                      const float* __restrict__ emb1,
                      unsigned char* uncl, Scal* sc, int first) {
  int done = sc->done;
  if (!first && !done) {                      // deferred count update for t-1
    int psum = sc->psum;
    float ratio = (float)sc->inter / (float)(psum > 1 ? psum : 1);
    if ((psum > MIN_INST_PIXEL) && (ratio > INST_RATIO)) sc->count += 1;
  }
  if (sc->sum_uncl <= MIN_PIXEL) done = 1;
  unsigned long long key = sc->argmax_key;
  unsigned sidx = 0xFFFFFFFFu - (unsigned)(key & 0xFFFFFFFFull);
  if (sidx >= HWN) sidx = 0;
  float score = __uint_as_float((unsigned)(key >> 32));
  if (!(score >= THRESHOLD)) done = 1;
  sc->done = done;
  sc->c0 = emb0[sidx];
  sc->c1 = emb1[sidx];
  sc->s0 = expf(pred[2u * HWN + sidx] * 10.0f);
  sc->s1 = expf(pred[3u * HWN + sidx] * 10.0f);
  uncl[sidx] = 0;                             // unclustered_s (dead state if done)
  sc->psum = 0; sc->inter = 0;                // for this iteration's k_dist
  sc->argmax_key = 0ull; sc->sum_uncl = 0;    // for next iteration's k_ar
}

// Distance pass: TDM-staged emb tiles in LDS, proposal mask, WMMA-based
// wave32 reductions of psum/inter.
__global__ __launch_bounds__(NT) void k_dist(const float* __restrict__ emb0,
                                             const float* __restrict__ emb1,
                                             const float* __restrict__ seedv,
                                             const unsigned char* __restrict__ uncl,
                                             unsigned char* __restrict__ prop,
                                             Scal* sc) {
  __shared__ float smem[2 * TILE];
  __shared__ float wsum[NT / 32][2];
  if (sc->done) return;                      // uniform across grid
  int base = blockIdx.x * TILE;
  if (threadIdx.x < 32) {                    // wave 0 drives the Tensor Data Mover
    unsigned l0 = (unsigned)(uintptr_t)&smem[0];
    unsigned l1 = (unsigned)(uintptr_t)&smem[TILE];
    tdm_load_f32_1d(emb0 + base, l0, TILE);
    tdm_load_f32_1d(emb1 + base, l1, TILE);
    __builtin_amdgcn_s_wait_tensorcnt(0);
  }
  __syncthreads();
  float c0 = sc->c0, c1 = sc->c1, s0 = sc->s0, s1 = sc->s1;
  int mp = 0, mi = 0;
  for (int k = threadIdx.x; k < TILE; k += NT) {
    int j = base + k;
    float d0 = smem[k] - c0;
    float d1 = smem[TILE + k] - c1;
    float q = d0 * d0 * s0 + d1 * d1 * s1;
    int v = seedv[j] > MIN_SEED_THRESH;
    int p = (q < DIST_LN2) & v;              // exp(-q) > 0.5
    prop[j] = (unsigned char)p;
    mp += p;
    mi += p & (int)uncl[j];
  }
  // wave32 sums on the matrix pipe (counts <= TILE are exact in f32)
  float tp = wave_sum_wmma((float)mp);
  float ti = wave_sum_wmma((float)mi);
  int wid  = threadIdx.x >> 5;
  int lane = threadIdx.x & 31;
  if (lane == 0) { wsum[wid][0] = tp; wsum[wid][1] = ti; }
  __syncthreads();
  if (threadIdx.x == 0) {
    float sp = 0.0f, si = 0.0f;
    #pragma unroll
    for (int wq = 0; wq < NT / 32; ++wq) { sp += wsum[wq][0]; si += wsum[wq][1]; }
    atomicAdd(&sc->psum, (int)sp);
    atomicAdd(&sc->inter, (int)si);
  }
}

// Final: label histogram.
__global__ __launch_bounds__(NT) void k_hist(const int* __restrict__ labels, Scal* sc) {
  __shared__ int h[68];
  if (threadIdx.x < 68) h[threadIdx.x] = 0;
  __syncthreads();
  int base = blockIdx.x * (NT * GRAIN);
  #pragma unroll
  for (int k = 0; k < GRAIN; ++k) {
    int j = base + k * NT + threadIdx.x;
    int l = labels[j];
    if ((unsigned)l > 65u) l = 0;
    atomicAdd(&h[l], 1);
  }
  __syncthreads();
  if (threadIdx.x < 68 && h[threadIdx.x]) atomicAdd(&sc->counts[threadIdx.x], h[threadIdx.x]);
}

// Final: zero labels of instances smaller than MIN_INST_PIXEL.
__global__ __launch_bounds__(NT) void k_final(int* __restrict__ labels, const Scal* sc) {
  int base = blockIdx.x * (NT * GRAIN);
  #pragma unroll
  for (int k = 0; k < GRAIN; ++k) {
    int j = base + k * NT + threadIdx.x;
    int l = labels[j];
    if (l > 0 && sc->counts[l] < MIN_INST_PIXEL) l = 0;
    labels[j] = l;
  }
}

// ---------------- host-side launch sequence ----------------
extern "C" void kernel_launch(void* const* d_in, const int* in_sizes, int n_in,
                              void* d_out, int out_size, void* d_ws, size_t ws_size,
                              hipStream_t stream) {
  (void)in_sizes; (void)n_in; (void)out_size; (void)ws_size;
  const float* pred = (const float*)d_in[0];     // [1,7,1024,2048] f32
  char* ws = (char*)d_ws;
  float* emb0  = (float*)ws;                     // HWN f32
  float* emb1  = emb0 + HWN;                     // HWN f32
  float* seedv = emb0 + 2u * HWN;                // HWN f32
  unsigned char* uncl = (unsigned char*)(ws + (size_t)12 * HWN);  // HWN u8
  unsigned char* prop = uncl + HWN;                               // HWN u8
  Scal* sc = (Scal*)(ws + (size_t)14 * HWN);
  int* labels = (int*)d_out;                     // output int32 labels [1,H,W]

  k_zero <<<1, NT, 0, stream>>>(sc);
  k_init <<<HWN / NT, NT, 0, stream>>>(pred, emb0, emb1, seedv, uncl, labels, sc);
  k_start<<<1, 1, 0, stream>>>(sc);

  for (int it = 0; it < MAX_ITERS; ++it) {
    k_ar  <<<HWN / (NT * GRAIN), NT, 0, stream>>>(seedv, uncl, prop, labels, sc, it == 0);
    k_mid <<<1, 1, 0, stream>>>(pred, emb0, emb1, uncl, sc, it == 0);
    k_dist<<<HWN / TILE, NT, 0, stream>>>(emb0, emb1, seedv, uncl, prop, sc);
  }
  // apply iteration 63's proposal
  k_ar  <<<HWN / (NT * GRAIN), NT, 0, stream>>>(seedv, uncl, prop, labels, sc, 0);

  k_hist <<<HWN / (NT * GRAIN), NT, 0, stream>>>(labels, sc);
  k_final<<<HWN / (NT * GRAIN), NT, 0, stream>>>(labels, sc);
}